// CachePBE_13554916786511
// MI455X (gfx1250) — compile-verified
//
#include <hip/hip_runtime.h>
#include <math.h>

// CDNA5 (gfx1250) wave32 WMMA types
typedef float v2f __attribute__((ext_vector_type(2)));
typedef float v8f __attribute__((ext_vector_type(8)));

#define N_Q     2048
#define M_BUF   131072
#define DDIM    64
#define KTOP    10
#define KP1     11                       // reference keeps k+1 then drops self-match
#define NSLAB   64                       // x rows per workgroup
#define NXT     (NSLAB / 16)             // 4 x-tiles of 16 columns
#define WAVES   8
#define BLOCK   (WAVES * 32)             // 256 threads = 8 wave32
#define MCHUNKS 8
#define TILES_PER_CHUNK ((M_BUF / 16) / MCHUNKS)   // 1024 buf tiles per WG
#define PLANE_STRIDE 160                 // words; 160 % 64 == 32 -> the two half-wave
                                         // groups cover disjoint 32-bank halves (no ds conflicts)

// Kernel 1: for each (query-slab, M-chunk) compute the 11 smallest squared
// distances per query row using f32 WMMA for the -2*x.buf^T term.
__global__ __launch_bounds__(BLOCK)
void pbe_partial_topk(const float* __restrict__ x,
                      const float* __restrict__ buf,
                      float* __restrict__ ws)
{
  // 45,056 B. First 5,120 floats double as the x B-operand plane store during
  // the streaming loop; after a barrier the whole region is reused for merging.
  __shared__ float smem[NSLAB * 16 * KP1];

  const int slab  = blockIdx.x;          // [0, 32)
  const int chunk = blockIdx.y;          // [0, MCHUNKS)
  const int tid   = threadIdx.x;
  const int lane  = tid & 31;
  const int wave  = tid >> 5;
  const int half  = (lane >> 4) & 1;     // 0: lanes 0-15, 1: lanes 16-31
  const int l16   = lane & 15;

  // ---- stage x slab into LDS in exact WMMA B-operand layout -----------------
  // plane p (0..31) holds the pair (x[n][2p], x[n][2p+1]) for slab rows n=0..63.
  // For k-step kk the wave needs plane 2*kk+half at word (t*16+l16)*2, which is
  // precisely B[k0..k0+1][n] with k0 = 4*kk + 2*half = 2*(2*kk+half).
  {
    const float* xs = x + (size_t)slab * NSLAB * DDIM;
    for (int i = tid; i < NSLAB * 32; i += BLOCK) {
      const int n = i >> 5, p = i & 31;
      const v2f v = *(const v2f*)(xs + n * DDIM + 2 * p);
      smem[p * PLANE_STRIDE + n * 2 + 0] = v.x;
      smem[p * PLANE_STRIDE + n * 2 + 1] = v.y;
    }
  }
  __syncthreads();

  // ---- streaming top-(K+1) on s = |b|^2 - 2*x.b (adding |x|^2 is a per-lane
  // constant here, hence rank-invariant; applied after the stream) ------------
  float lst[NXT][KP1];                   // sorted ascending, all static indices -> VGPRs
  #pragma unroll
  for (int t = 0; t < NXT; ++t)
    #pragma unroll
    for (int j = 0; j < KP1; ++j) lst[t][j] = 3.0e38f;

  const int mbase0 = chunk * TILES_PER_CHUNK;

  for (int ti = wave; ti < TILES_PER_CHUNK; ti += WAVES) {
    const int    mrow = (mbase0 + ti) * 16 + l16;        // this lane's buf row (A matrix M index)
    const float* rp   = buf + (size_t)mrow * DDIM;

    // A operand (16x4 f32 layout): lane holds buf[mrow][k0], buf[mrow][k0+1],
    // k0 = 4*kk + 2*half. Fold |b|^2 into the same pass.
    v2f   aP[16];
    float bnp = 0.f;
    #pragma unroll
    for (int kk = 0; kk < 16; ++kk) {
      aP[kk] = *(const v2f*)(rp + 4 * kk + 2 * half);
      bnp = fmaf(aP[kk].x, aP[kk].x, bnp);
      bnp = fmaf(aP[kk].y, aP[kk].y, bnp);
    }
    // prefetch this wave's next tile (128 candidate rows ahead) -> global_prefetch_b8
    __builtin_prefetch((const void*)(rp + 8 * 16 * DDIM), 0, 1);

    // combine the two K-halves: lane l now holds |b|^2 of tile row (l&15)
    const float bnAll = bnp + __shfl_xor(bnp, 16, 32);

    v8f acc[NXT];
    const v8f z = {0.f, 0.f, 0.f, 0.f, 0.f, 0.f, 0.f, 0.f};
    #pragma unroll
    for (int t = 0; t < NXT; ++t) acc[t] = z;

    // D = buf_tile(16xK) x x_slab^T(Kx16), K=64 via 16 steps of K=4, exact f32
    #pragma unroll
    for (int kk = 0; kk < 16; ++kk) {
      const v2f a = aP[kk];
      #pragma unroll
      for (int t = 0; t < NXT; ++t) {
        const v2f b = *(const v2f*)&smem[(2 * kk + half) * PLANE_STRIDE + (t * 16 + l16) * 2];
        acc[t] = __builtin_amdgcn_wmma_f32_16x16x4_f32(
            /*neg_a=*/false, a, /*neg_b=*/false, b,
            /*c_mod=*/(short)0, acc[t], /*reuse_a=*/false, /*reuse_b=*/false);
      }
    }

    // broadcast |b|^2 of the 8 tile rows this lane owns in the C layout
    // (VGPR g of lane l is row g + 8*(l>=16), column l&15)
    float bn8[8];
    #pragma unroll
    for (int g = 0; g < 8; ++g)
      bn8[g] = __shfl(bnAll, g + half * 8, 32);

    // branchless sorted insert (min/max bubble) — stays in registers
    #pragma unroll
    for (int t = 0; t < NXT; ++t) {
      #pragma unroll
      for (int g = 0; g < 8; ++g) {
        const float s = fmaf(acc[t][g], -2.0f, bn8[g]);
        if (s < lst[t][KP1 - 1]) {
          float cur = s;
          #pragma unroll
          for (int j = 0; j < KP1; ++j) {
            const float o  = lst[t][j];
            const float mn = fminf(cur, o);
            const float mx = fmaxf(cur, o);
            lst[t][j] = mn;
            cur = mx;
          }
        }
      }
    }
  }

  __syncthreads();   // x planes dead; reuse smem as the list-merge arena

  // write partial lists: row r = t*16+l16 in [0,64), list id = wave*2+half in [0,16)
  #pragma unroll
  for (int t = 0; t < NXT; ++t) {
    const int r  = t * 16 + l16;
    const int li = wave * 2 + half;
    #pragma unroll
    for (int j = 0; j < KP1; ++j)
      smem[(r * 16 + li) * KP1 + j] = lst[t][j];
  }
  __syncthreads();

  // one thread per x-row: 16 lists x 11 = 176 candidates -> 11 smallest d^2
  if (tid < NSLAB) {
    const int r  = tid;
    const int rg = slab * NSLAB + r;

    const float* xr = x + (size_t)rg * DDIM;
    float xn = 0.f;
    #pragma unroll
    for (int i = 0; i < DDIM; ++i) xn = fmaf(xr[i], xr[i], xn);

    float* seg = smem + r * 16 * KP1;
    float* wsp = ws + ((size_t)rg * MCHUNKS + chunk) * KP1;
    for (int j = 0; j < KP1; ++j) {
      float mn = 3.0e38f; int mi = 0;
      for (int i = 0; i < 16 * KP1; ++i) {
        const float v = seg[i];
        if (v < mn) { mn = v; mi = i; }
      }
      seg[mi] = 3.4e38f;                         // remove and emit ascending
      wsp[j]  = fmaxf(0.f, xn + mn);             // clamp matches reference
    }
  }
}

// Kernel 2: merge the MCHUNKS partial results per row, drop the self-match,
// then sqrt/mean/log1p.
__global__ __launch_bounds__(256)
void pbe_finalize(const float* __restrict__ ws, float* __restrict__ out)
{
  const int r = blockIdx.x * blockDim.x + threadIdx.x;
  if (r >= N_Q) return;
  const float* v = ws + (size_t)r * MCHUNKS * KP1;   // 88 candidates, each chunk ascending

  float prevV = -1.f;  // d^2 >= 0, so this admits everything on the first pass
  int   prevI = -1;
  float sum   = 0.f;
  for (int rank = 0; rank < KP1; ++rank) {
    float mn = 3.4e38f; int mi = -1;
    for (int i = 0; i < MCHUNKS * KP1; ++i) {
      const float vi = v[i];
      const bool gtPrev = (vi > prevV) || (vi == prevV && i > prevI);  // tie-safe strict order
      if (gtPrev && vi < mn) { mn = vi; mi = i; }
    }
    prevV = mn; prevI = mi;
    if (rank >= 1) sum += sqrtf(mn);                 // rank 0 is the exact self-match, dropped
  }
  out[r] = log1pf(sum * (1.0f / KTOP));
}

extern "C" void kernel_launch(void* const* d_in, const int* in_sizes, int n_in,
                              void* d_out, int out_size, void* d_ws, size_t ws_size,
                              hipStream_t stream)
{
  (void)in_sizes; (void)n_in; (void)out_size; (void)ws_size;
  const float* x   = (const float*)d_in[0];
  const float* buf = (const float*)d_in[1];
  float*       out = (float*)d_out;
  float*       ws  = (float*)d_ws;      // needs 2048*8*11*4 = 704 KiB of scratch

  dim3 grid1(N_Q / NSLAB, MCHUNKS);     // 32 slabs x 8 M-chunks = 256 workgroups
  pbe_partial_topk<<<grid1, BLOCK, 0, stream>>>(x, buf, ws);
  pbe_finalize<<<(N_Q + 255) / 256, 256, 0, stream>>>(ws, out);
}